// AGG_NET_42339787604899
// MI455X (gfx1250) — compile-verified
//
#include <hip/hip_runtime.h>
#include <hip/hip_bf16.h>

typedef __attribute__((ext_vector_type(2))) float v2f;
typedef __attribute__((ext_vector_type(8))) float v8f;

#define D 64
#define TILE_M 16
#define WAVES_PER_BLOCK 8

// --------------------------------------------------------------------------
// Dense GEMM: H[row, :] = X[row, :] @ W  (row-major, D=64)
// One wave32 computes a 16x64 output tile via V_WMMA_F32_16X16X4_F32.
// W is staged into LDS K-pair-interleaved so each lane's B fragment
// (two floats, adjacent K rows, same column) is one aligned ds_load_b64.
//   sWp[(k/2)*128 + n*2 + (k&1)] = W[k][n]
// --------------------------------------------------------------------------
__global__ __launch_bounds__(256) void gcn_gemm64(const float* __restrict__ X,
                                                  const float* __restrict__ W,
                                                  float* __restrict__ H,
                                                  int nRows) {
    __shared__ float sWp[D * D];   // 16 KB, permuted layout

    // cooperative permuted staging of W (256 threads * 16 elements)
    const int tid = threadIdx.x;
#pragma unroll
    for (int i = 0; i < (D * D) / 256; ++i) {
        const int idx = tid + i * 256;      // flat row-major index: k*64 + n
        const int k = idx >> 6;
        const int n = idx & 63;
        sWp[(k >> 1) * 128 + n * 2 + (k & 1)] = W[idx];
    }
    __syncthreads();

    const int wave = tid >> 5;            // 0..7
    const int lane = tid & 31;
    const int rowTile = blockIdx.x * WAVES_PER_BLOCK + wave;
    const int nTiles = nRows / TILE_M;    // N divisible by 16 (100000/16 = 6250)
    if (rowTile >= nTiles) return;        // wave-uniform exit: EXEC stays all-1s

    const int row0 = rowTile * TILE_M;
    const int r    = lane & 15;           // M index (A) / N-within-tile (B, C/D)
    const int kh   = lane >> 4;           // K half-select (0: K+{0,1}, 1: K+{2,3})

    v8f acc0 = {}, acc1 = {}, acc2 = {}, acc3 = {};

    const float* xrow = X + (size_t)(row0 + r) * D;

#pragma unroll
    for (int k0 = 0; k0 < D; k0 += 4) {
        const int ka = k0 + 2 * kh;       // this lane's first K
        // A fragment: 16x4 fp32, lane r holds row M=r, Ks {ka, ka+1}
        v2f a = *(const v2f*)(xrow + ka);
        // B fragments: one ds_load_b64 per 16-col tile
        const float* bbase = sWp + ((k0 >> 1) + kh) * 128;
#pragma unroll
        for (int n = 0; n < 4; ++n) {
            const v2f b = *(const v2f*)(bbase + (n * 16 + r) * 2);
            v8f c = (n == 0) ? acc0 : (n == 1) ? acc1 : (n == 2) ? acc2 : acc3;
            c = __builtin_amdgcn_wmma_f32_16x16x4_f32(
                    /*neg_a=*/false, a, /*neg_b=*/false, b,
                    /*c_mod=*/(short)0, c, /*reuse_a=*/false, /*reuse_b=*/false);
            if (n == 0) acc0 = c; else if (n == 1) acc1 = c;
            else if (n == 2) acc2 = c; else acc3 = c;
        }
    }

    // C/D layout: VGPR i, lanes 0-15 -> (M=i, N=lane); lanes 16-31 -> (M=i+8)
#pragma unroll
    for (int i = 0; i < 8; ++i) {
        float* orow = H + (size_t)(row0 + i + 8 * kh) * D + r;
        orow[0]  = acc0[i];
        orow[16] = acc1[i];
        orow[32] = acc2[i];
        orow[48] = acc3[i];
    }
}

// --------------------------------------------------------------------------
// Zero the aggregation target (float4-wide)
// --------------------------------------------------------------------------
__global__ void gcn_zero4(float4* __restrict__ p, int n4) {
    int i = blockIdx.x * blockDim.x + threadIdx.x;
    if (i < n4) p[i] = make_float4(0.f, 0.f, 0.f, 0.f);
}

// --------------------------------------------------------------------------
// Scatter-add: OUT[dst[e], :] += H[src[e], :]
// 16 threads per edge, each owning a float4 slice of the 64-wide feature row.
// Fully coalesced 256B gather; hardware global_atomic_add_f32 scatter
// (h and the 25.6MB target are L2-resident on the 192MB L2).
// --------------------------------------------------------------------------
__global__ __launch_bounds__(256) void gcn_scatter(const float* __restrict__ H,
                                                   const int* __restrict__ src,
                                                   const int* __restrict__ dst,
                                                   float* __restrict__ OUT,
                                                   int nEdges) {
    const int tid = blockIdx.x * blockDim.x + threadIdx.x;
    const int e = tid >> 4;
    if (e >= nEdges) return;
    const int d4 = (tid & 15) * 4;

    const int s = src[e];
    const int t = dst[e];

    const float4 v = *(const float4*)(H + (size_t)s * D + d4);
    float* o = OUT + (size_t)t * D + d4;
    unsafeAtomicAdd(o + 0, v.x);
    unsafeAtomicAdd(o + 1, v.y);
    unsafeAtomicAdd(o + 2, v.z);
    unsafeAtomicAdd(o + 3, v.w);
}

extern "C" void kernel_launch(void* const* d_in, const int* in_sizes, int n_in,
                              void* d_out, int out_size, void* d_ws, size_t ws_size,
                              hipStream_t stream) {
    const float* x   = (const float*)d_in[0];
    const int*   ei  = (const int*)d_in[1];     // [2, E] flat: src then dst
    const float* W0  = (const float*)d_in[2];
    const float* W1  = (const float*)d_in[3];
    const float* W2  = (const float*)d_in[4];

    const int N = in_sizes[0] / D;              // 100000
    const int E = in_sizes[1] / 2;              // 1200000
    const int* src = ei;
    const int* dst = ei + E;

    float* h   = (float*)d_ws;                  // N*D floats scratch (25.6 MB)
    float* out = (float*)d_out;                 // aggregation target / layer input

    const int nTiles     = N / TILE_M;
    const int gemmBlocks = (nTiles + WAVES_PER_BLOCK - 1) / WAVES_PER_BLOCK;
    const int n4         = (N * D) / 4;
    const int zeroBlocks = (n4 + 255) / 256;
    const long long sThreads = (long long)E * 16;
    const int scatBlocks = (int)((sThreads + 255) / 256);

    const float* Ws[3] = {W0, W1, W2};
    const float* cur = x;
    for (int layer = 0; layer < 3; ++layer) {
        // h = cur @ W
        gcn_gemm64<<<gemmBlocks, 256, 0, stream>>>(cur, Ws[layer], h, N);
        // out = 0 (safe: the previous gemm already consumed `out` as input)
        gcn_zero4<<<zeroBlocks, 256, 0, stream>>>((float4*)out, n4);
        // out[dst] += h[src]
        gcn_scatter<<<scatBlocks, 256, 0, stream>>>(h, src, dst, out, E);
        cur = out;
    }
}